// CrossFusionMamba_3238405341875
// MI455X (gfx1250) — compile-verified
//
#include <hip/hip_runtime.h>
#include <math.h>

// ---- model dims (fixed by reference) ----
#define BB   64
#define LL   512
#define VDIM 64
#define IDIM 32
#define HH   256
#define DIN  512    // d_inner
#define DST  16     // d_state
#define DCV  4      // d_conv
#define DRK  16     // dt_rank
#define NLAY 2
#define XPD  48     // DR + 2*DS

typedef __attribute__((ext_vector_type(2))) float v2f;
typedef __attribute__((ext_vector_type(8))) float v8f;
typedef __attribute__((ext_vector_type(4))) int   v4i;

__device__ __forceinline__ float cf_act(float x, int act) {
  if (act == 1) return x > 0.f ? x : 0.f;                          // relu
  if (act == 2) return x / (1.f + __expf(-x));                     // silu
  if (act == 3) return (x > 20.f) ? x : log1pf(__expf(x));         // softplus
  return x;
}

// ---------------------------------------------------------------------------
// Async global -> LDS staging (gfx1250 GLOBAL_LOAD_ASYNC_TO_LDS_B128,
// tracked by ASYNCcnt) with compile-safe fallback to plain LDS stores.
// Probe result: builtin exists; arg0 = v4i addrspace(1)*, arg1 = LDS ptr.
// ---------------------------------------------------------------------------
#if defined(__gfx1250__) && __has_builtin(__builtin_amdgcn_global_load_async_to_lds_b128)
#define CF_ASYNC 1
#else
#define CF_ASYNC 0
#endif

__device__ __forceinline__ void cf_cp16(const float* __restrict__ g,
                                        float* __restrict__ l) {
#if CF_ASYNC
  __builtin_amdgcn_global_load_async_to_lds_b128(
      (__attribute__((address_space(1))) v4i*)g,
      (__attribute__((address_space(3))) v4i*)l,
      /*imm offset*/0, /*cpol*/0);
#else
  *(float4*)l = *(const float4*)g;
#endif
}

__device__ __forceinline__ void cf_async_wait() {
#if CF_ASYNC
#if __has_builtin(__builtin_amdgcn_s_wait_asynccnt)
  __builtin_amdgcn_s_wait_asynccnt(0);
#else
  asm volatile("s_wait_asynccnt 0x0" ::: "memory");
#endif
#endif
}

// ---------------------------------------------------------------------------
// LDS-tiled fp32 GEMM:  C[m,n] = act( sum_k A[m,k]*W[n,k] + bias[n] )
// Block = 128 threads (4 waves), macro-tile 64x64, BK=16, double-buffered LDS.
// Each wave owns a 32x32 sub-tile: 2x2 accumulators of V_WMMA_F32_16X16X4_F32.
// Per staged K-block a wave issues 16 WMMAs fed by ds_load_b64 fragments.
// ---------------------------------------------------------------------------
#define BMT 64
#define BNT 64
#define BKT 16

__global__ __launch_bounds__(128)
void cf_gemm_wmma_lds(const float* __restrict__ A, int lda,
                      const float* __restrict__ W, int ldw,
                      const float* __restrict__ bias,
                      float* __restrict__ C, int ldc,
                      int K, int act) {
  __shared__ __align__(16) float As[2][BMT * BKT];  // [row][k]
  __shared__ __align__(16) float Ws[2][BNT * BKT];  // [col][k]

  const int tid  = threadIdx.x;       // 0..127
  const int lane = tid & 31;
  const int wid  = tid >> 5;          // 0..3
  const int half = lane >> 4;         // 0|1
  const int lr   = lane & 15;
  const int wrow = (wid >> 1) * 32;   // 0|32
  const int wcol = (wid & 1) * 32;    // 0|32

  const int rowBase = blockIdx.y * BMT;
  const int colBase = blockIdx.x * BNT;

  // Cooperative staging: 64 rows x 16 floats = 256 float4 slots per matrix;
  // 128 threads move 2 slots each. slot s: row = s>>2, kq = (s&3)*4.
  auto stage = [&](int buf, int k0) {
#pragma unroll
    for (int r = 0; r < 2; ++r) {
      const int s   = tid + r * 128;
      const int row = s >> 2;
      const int kq  = (s & 3) * 4;
      cf_cp16(&A[(size_t)(rowBase + row) * lda + k0 + kq],
              &As[buf][row * BKT + kq]);
      cf_cp16(&W[(size_t)(colBase + row) * ldw + k0 + kq],
              &Ws[buf][row * BKT + kq]);
    }
  };

  v8f acc[2][2] = {};
  const int nkb = K / BKT;

  stage(0, 0);
  for (int kb = 0; kb < nkb; ++kb) {
    cf_async_wait();
    __syncthreads();                       // staged data visible; prior reads done
    if (kb + 1 < nkb) stage((kb + 1) & 1, (kb + 1) * BKT);

    const float* __restrict__ as = As[kb & 1];
    const float* __restrict__ ws = Ws[kb & 1];
#pragma unroll
    for (int k = 0; k < BKT; k += 4) {
      v2f ar[2], bc[2];
#pragma unroll
      for (int i = 0; i < 2; ++i) {
        const float2 t =
            *(const float2*)&as[(wrow + i * 16 + lr) * BKT + k + 2 * half];
        ar[i].x = t.x; ar[i].y = t.y;
      }
#pragma unroll
      for (int j = 0; j < 2; ++j) {
        const float2 t =
            *(const float2*)&ws[(wcol + j * 16 + lr) * BKT + k + 2 * half];
        bc[j].x = t.x; bc[j].y = t.y;
      }
#pragma unroll
      for (int i = 0; i < 2; ++i)
#pragma unroll
        for (int j = 0; j < 2; ++j)
          acc[i][j] = __builtin_amdgcn_wmma_f32_16x16x4_f32(
              false, ar[i], false, bc[j], (short)0, acc[i][j], false, false);
    }
  }

#pragma unroll
  for (int i = 0; i < 2; ++i)
#pragma unroll
    for (int j = 0; j < 2; ++j)
#pragma unroll
      for (int r = 0; r < 8; ++r) {
        const int row = rowBase + wrow + i * 16 + r + 8 * half;
        const int col = colBase + wcol + j * 16 + lr;
        float v = acc[i][j][r];
        if (bias) v += bias[col];
        C[(size_t)row * ldc + col] = cf_act(v, act);
      }
}

// ---------------------------------------------------------------------------
// Fallback direct-GEMM (one wave per 16x(16*NT) tile) for shapes the LDS
// kernel can't cover (x_proj: N=48).
// ---------------------------------------------------------------------------
template <int NT>
__global__ void cf_gemm_wmma(const float* __restrict__ A, int lda,
                             const float* __restrict__ W, int ldw,
                             const float* __restrict__ bias,
                             float* __restrict__ C, int ldc,
                             int K, int act) {
  const int lane = threadIdx.x;
  const int half = lane >> 4;
  const int lr   = lane & 15;
  const int row0 = blockIdx.y * 16;
  const int col0 = blockIdx.x * (16 * NT);

  v8f acc[NT] = {};
  const float* __restrict__ arow = A + (size_t)(row0 + lr) * lda;
  for (int k = 0; k < K; k += 4) {
    v2f a;
    a.x = arow[k + 2 * half + 0];
    a.y = arow[k + 2 * half + 1];
#pragma unroll
    for (int j = 0; j < NT; ++j) {
      const float* __restrict__ wrow = W + (size_t)(col0 + 16 * j + lr) * ldw;
      v2f b;
      b.x = wrow[k + 2 * half + 0];
      b.y = wrow[k + 2 * half + 1];
      acc[j] = __builtin_amdgcn_wmma_f32_16x16x4_f32(
          false, a, false, b, (short)0, acc[j], false, false);
    }
  }
#pragma unroll
  for (int j = 0; j < NT; ++j)
#pragma unroll
    for (int r = 0; r < 8; ++r) {
      const int row = row0 + r + 8 * half;
      const int col = col0 + 16 * j + lr;
      float v = acc[j][r];
      if (bias) v += bias[col];
      C[(size_t)row * ldc + col] = cf_act(v, act);
    }
}

// ---- LayerNorm over last dim (width HH), one block per row ----
__global__ void cf_ln(const float* __restrict__ in, const float* __restrict__ w,
                      const float* __restrict__ b, float* __restrict__ out) {
  __shared__ float red[HH];
  const int row = blockIdx.x, tid = threadIdx.x;
  const float v = in[(size_t)row * HH + tid];
  red[tid] = v;
  __syncthreads();
  for (int s = HH / 2; s > 0; s >>= 1) {
    if (tid < s) red[tid] += red[tid + s];
    __syncthreads();
  }
  const float mu = red[0] * (1.f / HH);
  __syncthreads();
  const float d = v - mu;
  red[tid] = d * d;
  __syncthreads();
  for (int s = HH / 2; s > 0; s >>= 1) {
    if (tid < s) red[tid] += red[tid + s];
    __syncthreads();
  }
  const float var = red[0] * (1.f / HH);
  out[(size_t)row * HH + tid] = d * rsqrtf(var + 1e-5f) * w[tid] + b[tid];
}

// ---- causal depthwise conv1d (DC=4) + SiLU; u read from xz[:, :DIN] ----
__global__ void cf_conv_silu(const float* __restrict__ xz,
                             const float* __restrict__ cw,
                             const float* __restrict__ cb,
                             float* __restrict__ u) {
  const int total = BB * LL * DIN;
  for (int idx = blockIdx.x * blockDim.x + threadIdx.x; idx < total;
       idx += gridDim.x * blockDim.x) {
    const int d = idx % DIN;
    const int l = (idx / DIN) % LL;
    const int b = idx / (DIN * LL);
    float acc = cb[d];
#pragma unroll
    for (int k = 0; k < DCV; ++k) {
      const int t = l + k - (DCV - 1);
      if (t >= 0)
        acc += cw[d * DCV + k] * xz[(size_t)(b * LL + t) * (2 * DIN) + d];
    }
    u[idx] = acc / (1.f + __expf(-acc));
  }
}

// ---- selective scan: grid=B, block=DIN; h[DS] and A row live in registers ----
__global__ void cf_scan(const float* __restrict__ u, const float* __restrict__ dt,
                        const float* __restrict__ xdbl,  // [B*L, 48]; Bc@16, Cc@32
                        const float* __restrict__ Alog, const float* __restrict__ Dp,
                        float* __restrict__ y) {
  __shared__ float sB[DST], sC[DST];
  const int b = blockIdx.x;
  const int d = threadIdx.x;
  float A[DST], h[DST];
#pragma unroll
  for (int n = 0; n < DST; ++n) {
    A[n] = -__expf(Alog[d * DST + n]);
    h[n] = 0.f;
  }
  const float Dd = Dp[d];
  for (int t = 0; t < LL; ++t) {
    __syncthreads();
    if (d < 32) {
      const float* p = xdbl + (size_t)(b * LL + t) * XPD + DRK;  // Bc then Cc
      if (d < DST) sB[d] = p[d];
      else         sC[d - DST] = p[d];
    }
    __syncthreads();
    const size_t off = (size_t)(b * LL + t) * DIN + d;
    const float dtv = dt[off];
    const float uv  = u[off];
    const float du  = dtv * uv;
    float acc = 0.f;
#pragma unroll
    for (int n = 0; n < DST; ++n) {
      h[n] = __expf(dtv * A[n]) * h[n] + du * sB[n];
      acc += h[n] * sC[n];
    }
    y[off] = acc + uv * Dd;
  }
}

// ---- y *= silu(z); z = xz[:, DIN:2*DIN] ----
__global__ void cf_gate(float* __restrict__ y, const float* __restrict__ xz) {
  const int total = BB * LL * DIN;
  for (int idx = blockIdx.x * blockDim.x + threadIdx.x; idx < total;
       idx += gridDim.x * blockDim.x) {
    const int d = idx % DIN;
    const size_t rl = idx / DIN;
    const float z = xz[rl * (2 * DIN) + DIN + d];
    y[idx] *= z / (1.f + __expf(-z));
  }
}

// ---- attention pool over time: grid=B, block=HH ----
__global__ void cf_pool(const float* __restrict__ x, const float* __restrict__ pw,
                        const float* __restrict__ pb, float* __restrict__ v) {
  __shared__ float sc[LL];
  __shared__ float red[HH];
  const int b = blockIdx.x, tid = threadIdx.x;
  for (int l = tid; l < LL; l += HH) {
    const float* xr = x + (size_t)(b * LL + l) * HH;
    float s = 0.f;
    for (int h = 0; h < HH; ++h) s += xr[h] * pw[h];
    sc[l] = s + pb[0];
  }
  __syncthreads();
  red[tid] = fmaxf(sc[tid], sc[tid + HH]);
  __syncthreads();
  for (int s = HH / 2; s > 0; s >>= 1) {
    if (tid < s) red[tid] = fmaxf(red[tid], red[tid + s]);
    __syncthreads();
  }
  const float mx = red[0];
  __syncthreads();
  for (int l = tid; l < LL; l += HH) sc[l] = __expf(sc[l] - mx);
  __syncthreads();
  red[tid] = sc[tid] + sc[tid + HH];
  __syncthreads();
  for (int s = HH / 2; s > 0; s >>= 1) {
    if (tid < s) red[tid] += red[tid + s];
    __syncthreads();
  }
  const float Z = red[0];
  float acc = 0.f;
  for (int l = 0; l < LL; ++l) acc += sc[l] * x[(size_t)(b * LL + l) * HH + tid];
  v[b * HH + tid] = acc / Z;
}

// ---- fusion concat: f = [v, i, v*i] ----
__global__ void cf_fuse(const float* __restrict__ v, const float* __restrict__ img,
                        float* __restrict__ f) {
  const int idx = blockIdx.x * blockDim.x + threadIdx.x;
  if (idx >= BB * 3 * HH) return;
  const int b = idx / (3 * HH);
  const int c = idx % (3 * HH);
  float out;
  if (c < HH)            out = v[b * HH + c];
  else if (c < 2 * HH)   out = img[b * HH + (c - HH)];
  else                   out = v[b * HH + (c - 2 * HH)] * img[b * HH + (c - 2 * HH)];
  f[idx] = out;
}

// ---- final head: out[b] = dot(h1[b,:], w) + bias ----
__global__ void cf_head2(const float* __restrict__ h1, const float* __restrict__ w,
                         const float* __restrict__ bias, float* __restrict__ out) {
  __shared__ float red[HH];
  const int b = blockIdx.x, tid = threadIdx.x;
  red[tid] = h1[b * HH + tid] * w[tid];
  __syncthreads();
  for (int s = HH / 2; s > 0; s >>= 1) {
    if (tid < s) red[tid] += red[tid + s];
    __syncthreads();
  }
  if (tid == 0) out[b] = red[0] + bias[0];
}

// ---------------------------------------------------------------------------
static void launch_gemm(const float* A, int lda, const float* W, int ldw,
                        const float* bias, float* C, int ldc,
                        int M, int N, int K, int act, hipStream_t s) {
  if (M % BMT == 0 && N % BNT == 0 && K % BKT == 0) {
    dim3 g(N / BNT, M / BMT);
    cf_gemm_wmma_lds<<<g, 128, 0, s>>>(A, lda, W, ldw, bias, C, ldc, K, act);
  } else if (N % 48 == 0) {
    dim3 g(N / 48, M / 16);
    cf_gemm_wmma<3><<<g, 32, 0, s>>>(A, lda, W, ldw, bias, C, ldc, K, act);
  } else {
    dim3 g(N / 16, M / 16);
    cf_gemm_wmma<1><<<g, 32, 0, s>>>(A, lda, W, ldw, bias, C, ldc, K, act);
  }
}

extern "C" void kernel_launch(void* const* d_in, const int* in_sizes, int n_in,
                              void* d_out, int out_size, void* d_ws, size_t ws_size,
                              hipStream_t stream) {
  (void)in_sizes; (void)n_in; (void)out_size; (void)ws_size;
  const float* xv        = (const float*)d_in[0];
  const float* xi        = (const float*)d_in[1];
  const float* vent_in_w = (const float*)d_in[2];
  const float* vent_in_b = (const float*)d_in[3];
  const float* vent_ln_w = (const float*)d_in[4];
  const float* vent_ln_b = (const float*)d_in[5];
  const float* m_in_w    = (const float*)d_in[6];
  const float* m_conv_w  = (const float*)d_in[7];
  const float* m_conv_b  = (const float*)d_in[8];
  const float* m_xproj_w = (const float*)d_in[9];
  const float* m_dt_w    = (const float*)d_in[10];
  const float* m_dt_b    = (const float*)d_in[11];
  const float* m_Alog    = (const float*)d_in[12];
  const float* m_D       = (const float*)d_in[13];
  const float* m_out_w   = (const float*)d_in[14];
  const float* m_ln_w    = (const float*)d_in[15];
  const float* m_ln_b    = (const float*)d_in[16];
  const float* pool_w    = (const float*)d_in[17];
  const float* pool_b    = (const float*)d_in[18];
  const float* img_w1    = (const float*)d_in[19];
  const float* img_b1    = (const float*)d_in[20];
  const float* img_w2    = (const float*)d_in[21];
  const float* img_b2    = (const float*)d_in[22];
  const float* head_w1   = (const float*)d_in[23];
  const float* head_b1   = (const float*)d_in[24];
  const float* head_w2   = (const float*)d_in[25];
  const float* head_b2   = (const float*)d_in[26];
  float* outp = (float*)d_out;

  // workspace carve (floats)
  float* ws = (float*)d_ws;
  size_t off = 0;
  auto carve = [&](size_t n) { float* p = ws + off; off += n; return p; };
  const size_t M = (size_t)BB * LL;  // 32768 tokens
  float* X    = carve(M * HH);
  float* TMP  = carve(M * HH);
  float* XZ   = carve(M * 2 * DIN);
  float* U    = carve(M * DIN);
  float* DT   = carve(M * DIN);
  float* XDBL = carve(M * XPD);
  float* Y    = carve(M * DIN);
  float* V    = carve((size_t)BB * HH);
  float* IH   = carve((size_t)BB * HH);
  float* IMG  = carve((size_t)BB * HH);
  float* F    = carve((size_t)BB * 3 * HH);
  float* H1   = carve((size_t)BB * HH);

  const int EW_BLOCKS = 8192;

  // vent_in: Linear + LN
  launch_gemm(xv, VDIM, vent_in_w, VDIM, vent_in_b, TMP, HH,
              (int)M, HH, VDIM, /*act*/0, stream);
  cf_ln<<<(int)M, HH, 0, stream>>>(TMP, vent_ln_w, vent_ln_b, X);

  for (int l = 0; l < NLAY; ++l) {
    const float* in_w  = m_in_w    + (size_t)l * (2 * DIN) * HH;
    const float* cw    = m_conv_w  + (size_t)l * DIN * DCV;
    const float* cb    = m_conv_b  + (size_t)l * DIN;
    const float* xp_w  = m_xproj_w + (size_t)l * XPD * DIN;
    const float* dt_w  = m_dt_w    + (size_t)l * DIN * DRK;
    const float* dt_b  = m_dt_b    + (size_t)l * DIN;
    const float* Alog  = m_Alog    + (size_t)l * DIN * DST;
    const float* Dp    = m_D       + (size_t)l * DIN;
    const float* out_w = m_out_w   + (size_t)l * HH * DIN;
    const float* ln_w  = m_ln_w    + (size_t)l * HH;
    const float* ln_b  = m_ln_b    + (size_t)l * HH;

    // xz = x @ in_w.T   [M, 1024]
    launch_gemm(X, HH, in_w, HH, nullptr, XZ, 2 * DIN, (int)M, 2 * DIN, HH, 0, stream);
    // u = silu(causal depthwise conv(xz[:, :DIN]))
    cf_conv_silu<<<EW_BLOCKS, 256, 0, stream>>>(XZ, cw, cb, U);
    // xdbl = u @ xp_w.T  [M, 48]  (dt|Bc|Cc)
    launch_gemm(U, DIN, xp_w, DIN, nullptr, XDBL, XPD, (int)M, XPD, DIN, 0, stream);
    // dt = softplus(xdbl[:, :16] @ dt_w.T + dt_b)  [M, 512]
    launch_gemm(XDBL, XPD, dt_w, DRK, dt_b, DT, DIN, (int)M, DIN, DRK, /*softplus*/3, stream);
    // selective scan
    cf_scan<<<BB, DIN, 0, stream>>>(U, DT, XDBL, Alog, Dp, Y);
    // y *= silu(z)
    cf_gate<<<EW_BLOCKS, 256, 0, stream>>>(Y, XZ);
    // out = LN(y @ out_w.T)
    launch_gemm(Y, DIN, out_w, DIN, nullptr, TMP, HH, (int)M, HH, DIN, 0, stream);
    cf_ln<<<(int)M, HH, 0, stream>>>(TMP, ln_w, ln_b, X);
  }

  // attention pool over time
  cf_pool<<<BB, HH, 0, stream>>>(X, pool_w, pool_b, V);
  // image branch MLP (relu fused)
  launch_gemm(xi, IDIM, img_w1, IDIM, img_b1, IH, HH, BB, HH, IDIM, 1, stream);
  launch_gemm(IH, HH, img_w2, HH, img_b2, IMG, HH, BB, HH, HH, 1, stream);
  // fusion head
  cf_fuse<<<(BB * 3 * HH + 255) / 256, 256, 0, stream>>>(V, IMG, F);
  launch_gemm(F, 3 * HH, head_w1, 3 * HH, head_b1, H1, HH, BB, HH, 3 * HH, 1, stream);
  cf_head2<<<BB, HH, 0, stream>>>(H1, head_w2, head_b2, outp);
}